// SAGE_12077448036841
// MI455X (gfx1250) — compile-verified
//
#include <hip/hip_runtime.h>
#include <hip/hip_bf16.h>
#include <stdint.h>

typedef __attribute__((ext_vector_type(2))) float    v2f;
typedef __attribute__((ext_vector_type(8))) float    v8f;
typedef __attribute__((ext_vector_type(4))) uint32_t v4u32;
typedef __attribute__((ext_vector_type(8))) int      v8i32;
typedef __attribute__((ext_vector_type(4))) int      v4i32;

#define D 128
// LDS tile row stride in dwords: 128 data + 4 pad (TDM pad_amount=4) ->
// bank = (4*M + k) % 64, conflict-free across the wave, 8B alignment kept.
#define TROW 132
#define TILE_DW (16 * TROW)          // 2112 dwords per 16x128 tile (incl. pad)

// ---------------------------------------------------------------------------
// Edge scatter: one wave32 per edge; coalesced float4 gather of the source
// row, f32 atomic scatter-add into the L2-resident accumulator.
// ---------------------------------------------------------------------------
__global__ __launch_bounds__(256) void sage_scatter(
    const float* __restrict__ h, const int* __restrict__ src,
    const int* __restrict__ dst, float* __restrict__ agg,
    float* __restrict__ deg, int n_edges)
{
    int gid  = blockIdx.x * blockDim.x + threadIdx.x;
    int e    = gid >> 5;
    int lane = gid & 31;
    if (e >= n_edges) return;

    int s = src[e];
    int d = dst[e];

    const float4 v = ((const float4*)(h + (size_t)s * D))[lane];
    float* a = agg + (size_t)d * D + lane * 4;
    atomicAdd(a + 0, v.x);
    atomicAdd(a + 1, v.y);
    atomicAdd(a + 2, v.z);
    atomicAdd(a + 3, v.w);

    if (deg != nullptr && lane == 0) atomicAdd(&deg[d], 1.0f);
}

// ---------------------------------------------------------------------------
// Transpose the four 128x128 weight matrices into Wt (so B-fragments become
// single aligned b64 loads). Tiny one-shot kernel; Wt stays L2-resident.
// ---------------------------------------------------------------------------
__global__ __launch_bounds__(256) void transpose_weights(
    const float* __restrict__ W0, const float* __restrict__ W1,
    const float* __restrict__ W2, const float* __restrict__ W3,
    float* __restrict__ Wt)
{
    int idx = blockIdx.x * blockDim.x + threadIdx.x;   // 4 * 128 * 128 threads
    int m = idx >> 14;
    int r = idx & 16383;
    int n = r >> 7;
    int k = r & 127;
    const float* W = (m == 0) ? W0 : (m == 1) ? W1 : (m == 2) ? W2 : W3;
    Wt[(size_t)m * D * D + (size_t)n * D + k] = W[(size_t)k * D + n];
}

// ---------------------------------------------------------------------------
// Build a TDM descriptor for a 16x128 f32 tile (row-major, stride 128) and
// issue TENSOR_LOAD_TO_LDS.  D# layout per CDNA5 ISA.
// Toolchain here is the 6-arg builtin form:
//   (uint32x4 g0, int32x8 g1, int32x4 g2, int32x4 g3, int32x8 gx, i32 cpol)
// ---------------------------------------------------------------------------
__device__ __forceinline__ void tdm_load_tile(
    const float* gptr, uint32_t lds_byte_off, int rows_remaining)
{
    uint64_t ga = (uint64_t)(uintptr_t)gptr;

    v4u32 g0;
    g0.x = 1u;                                     // count=1, user descriptor
    g0.y = lds_byte_off;                           // LDS dest address
    g0.z = (uint32_t)ga;                           // global_addr[31:0]
    g0.w = ((uint32_t)(ga >> 32) & 0x01FFFFFFu)    // global_addr[56:32]
           | (2u << 30);                           // type=2 ("image")

    uint32_t td1 = (uint32_t)rows_remaining;       // OOB rows read as zero
    v8i32 g1;
    g1[0] = (int)((2u << 16)       // data_size = 4 bytes
                | (1u << 20)       // pad_enable
                | (6u << 22)       // pad_interval = 128 dwords (one row)
                | (3u << 25));     // pad_amount   = 4 dwords
    g1[1] = (int)(128u << 16);                     // tensor_dim0 = 128 (lo16)
    g1[2] = (int)((td1 & 0xFFFFu) << 16);          // tensor_dim0 hi=0 | td1 lo16
    g1[3] = (int)(((td1 >> 16) & 0xFFFFu)          // tensor_dim1 hi16
                | (128u << 16));                   // tile_dim0 = 128
    g1[4] = 16;                                    // tile_dim1 = 16, tile_dim2 = 0
    g1[5] = 128;                                   // tensor_dim0_stride lo32
    g1[6] = 0;                                     // stride hi | dim1_stride lo
    g1[7] = 0;

    v4i32 gz4 = {};                                // groups 2/3 unused (2-D tile)
    v8i32 gz8 = {};
    __builtin_amdgcn_tensor_load_to_lds(g0, g1, gz4, gz4, gz8, 0);
}

// ---------------------------------------------------------------------------
// Fused SAGE layer: out = relu(Hs @ Ws + (Hn/deg) @ Wn + b)
// Block = 8 waves; block owns 16 rows, wave w owns columns [16w,16w+16).
// A-tiles (Hs, Hn) staged into LDS by the Tensor Data Mover; f32 WMMA
// 16x16x4, K=128 in 32 steps, 2 WMMAs/step into one accumulator.
// ---------------------------------------------------------------------------
__global__ __launch_bounds__(256) void sage_layer(
    const float* __restrict__ Hs, const float* __restrict__ Hn,
    const float* __restrict__ deg,
    const float* __restrict__ WsT, const float* __restrict__ WnT,
    const float* __restrict__ bias, float* __restrict__ out, int n_nodes)
{
    __shared__ float tile[2 * TILE_DW];            // 16.5 KB: Hs tile | Hn tile

    const int lane = threadIdx.x & 31;
    const int wave = threadIdx.x >> 5;
    const int half = lane >> 4;
    const int mr   = lane & 15;
    const int m0   = blockIdx.x * 16;

    if (wave == 0) {
        uint32_t base = (uint32_t)(uintptr_t)&tile[0];
        int rem = n_nodes - m0;
        tdm_load_tile(Hs + (size_t)m0 * D, base,                      rem);
        tdm_load_tile(Hn + (size_t)m0 * D, base + TILE_DW * 4u,       rem);
        __builtin_amdgcn_s_wait_tensorcnt(0);
    }
    __syncthreads();

    int row = m0 + mr;
    row = row < n_nodes ? row : n_nodes - 1;       // branchless: EXEC all-ones
    const int col = wave * 16 + mr;
    const int kh  = half * 2;

    const float inv = __builtin_amdgcn_rcpf(fmaxf(deg[row], 1.0f));

    const float* ts = &tile[mr * TROW];
    const float* tn = &tile[TILE_DW + mr * TROW];
    const float* ws = WsT + ((size_t)col << 7);
    const float* wn = WnT + ((size_t)col << 7);

    v8f c = {};
    #pragma unroll
    for (int k0 = 0; k0 < D; k0 += 4) {
        const int k = k0 + kh;                     // even -> 8B-aligned
        v2f a_s = *(const v2f*)(ts + k);
        v2f a_n = *(const v2f*)(tn + k);
        a_n = a_n * inv;                           // mean aggregation
        v2f b_s = *(const v2f*)(ws + k);
        v2f b_n = *(const v2f*)(wn + k);
        c = __builtin_amdgcn_wmma_f32_16x16x4_f32(
                false, a_s, false, b_s, (short)0, c, false, false);
        c = __builtin_amdgcn_wmma_f32_16x16x4_f32(
                false, a_n, false, b_n, (short)0, c, false, false);
    }

    const float bb = bias[col];
    #pragma unroll
    for (int r = 0; r < 8; ++r) {
        const int m = m0 + r + 8 * half;
        if (m < n_nodes) {
            float v = c[r] + bb;
            out[(size_t)m * D + col] = fmaxf(v, 0.0f);
        }
    }
}

// ---------------------------------------------------------------------------
// Row L2-normalize: wave per row, float4 per lane, shfl_xor + v_rsq epilogue.
// ---------------------------------------------------------------------------
__global__ __launch_bounds__(256) void sage_normalize(
    const float* __restrict__ h, float* __restrict__ out, int n_nodes)
{
    int gid  = blockIdx.x * blockDim.x + threadIdx.x;
    int row  = gid >> 5;
    int lane = gid & 31;
    if (row >= n_nodes) return;

    const float4 v = ((const float4*)(h + (size_t)row * D))[lane];
    float s = v.x * v.x + v.y * v.y + v.z * v.z + v.w * v.w;
    #pragma unroll
    for (int off = 16; off > 0; off >>= 1) s += __shfl_xor(s, off, 32);

    const float scale = __builtin_amdgcn_rsqf(fmaxf(s, 1e-24f));
    float4 o;
    o.x = v.x * scale; o.y = v.y * scale; o.z = v.z * scale; o.w = v.w * scale;
    ((float4*)(out + (size_t)row * D))[lane] = o;
}

// ---------------------------------------------------------------------------
extern "C" void kernel_launch(void* const* d_in, const int* in_sizes, int n_in,
                              void* d_out, int out_size, void* d_ws, size_t ws_size,
                              hipStream_t stream)
{
    const float* x   = (const float*)d_in[0];
    const int*   src = (const int*)  d_in[1];
    const int*   dst = (const int*)  d_in[2];
    const float* Ws0 = (const float*)d_in[3];
    const float* Wn0 = (const float*)d_in[4];
    const float* b0  = (const float*)d_in[5];
    const float* Ws1 = (const float*)d_in[6];
    const float* Wn1 = (const float*)d_in[7];
    const float* b1  = (const float*)d_in[8];

    const int N = in_sizes[0] / D;
    const int E = in_sizes[1];

    // Workspace: agg | deg | h1 | h2 | Wt (4 transposed 128x128 matrices)
    float* agg = (float*)d_ws;
    float* deg = agg + (size_t)N * D;
    size_t degPad = ((size_t)N + 3) & ~(size_t)3;
    float* h1  = deg + degPad;
    float* h2  = h1 + (size_t)N * D;
    float* wt  = h2 + (size_t)N * D;
    float* WsT0 = wt + 0 * D * D;
    float* WnT0 = wt + 1 * D * D;
    float* WsT1 = wt + 2 * D * D;
    float* WnT1 = wt + 3 * D * D;

    const size_t aggBytes = (size_t)N * D * sizeof(float);

    const int scatterBlocks = (int)(((size_t)E * 32 + 255) / 256);
    const int gemmBlocks    = (N + 15) / 16;
    const int normBlocks    = (int)(((size_t)N * 32 + 255) / 256);

    // One-shot weight transposes (64 KB each; L2-resident afterwards)
    transpose_weights<<<(4 * D * D) / 256, 256, 0, stream>>>(Ws0, Wn0, Ws1, Wn1, wt);

    // ---- Layer 1 ----
    (void)hipMemsetAsync(agg, 0, aggBytes, stream);
    (void)hipMemsetAsync(deg, 0, degPad * sizeof(float), stream);
    sage_scatter<<<scatterBlocks, 256, 0, stream>>>(x, src, dst, agg, deg, E);
    sage_layer<<<gemmBlocks, 256, 0, stream>>>(x, agg, deg, WsT0, WnT0, b0, h1, N);

    // ---- Layer 2 (degrees unchanged) ----
    (void)hipMemsetAsync(agg, 0, aggBytes, stream);
    sage_scatter<<<scatterBlocks, 256, 0, stream>>>(h1, src, dst, agg, nullptr, E);
    sage_layer<<<gemmBlocks, 256, 0, stream>>>(h1, agg, deg, WsT1, WnT1, b1, h2, N);

    // ---- Final row L2 normalize ----
    sage_normalize<<<normBlocks, 256, 0, stream>>>(h2, (float*)d_out, N);
}